// TTLayer_47571057771201
// MI455X (gfx1250) — compile-verified
//
#include <hip/hip_runtime.h>
#include <hip/hip_bf16.h>

typedef float v2f __attribute__((ext_vector_type(2)));
typedef float v8f __attribute__((ext_vector_type(8)));

static __device__ __forceinline__ v8f wmma4(v8f c, v2f a, v2f b) {
  // D = A(16x4 f32) * B(4x16 f32) + C(16x16 f32)
  return __builtin_amdgcn_wmma_f32_16x16x4_f32(false, a, false, b, (short)0, c,
                                               false, false);
}

// One workgroup per (batch b, output mode o3). 256 threads = 8 wave32.
// LDS: single 512x64 f32 buffer (128 KB), reused (ping in place) across phases.
__global__ __launch_bounds__(256)
void tt_fused_kernel(const float* __restrict__ x,
                     const float* __restrict__ c0,   // (1,8,8,64)  idx=(i0*8+o0)*64+r1
                     const float* __restrict__ c1,   // (64,8,8,64) idx=r1*4096+i1*512+o1*64+r2
                     const float* __restrict__ c2,   // (64,8,8,64) idx=r2*4096+i2*512+o2*64+r3
                     const float* __restrict__ c3,   // (64,8,8,1)  idx=(r3*8+i3)*8+o3
                     const float* __restrict__ bias,
                     float* __restrict__ y)
{
  extern __shared__ float smem[];            // 512*64 floats = 128 KB

  const int tid  = threadIdx.x;
  const int wave = tid >> 5;                 // 0..7
  const int lane = tid & 31;
  const int lh   = lane >> 4;                // half-wave select
  const int lm   = lane & 15;
  const int b    = blockIdx.x >> 3;
  const int o3   = blockIdx.x & 7;

  const float* __restrict__ xrow = x + (size_t)b * 4096;
  const v8f vzero = {0.f,0.f,0.f,0.f,0.f,0.f,0.f,0.f};

  // ---------------- Phase 3: T3[i012=512][r3=64] = x(512x8) * W3(8x64) ----
  // W3[i3,r3] = c3[r3,i3,o3]. Wave w owns M-tiles 4w..4w+3, all 4 N-tiles.
  for (int nt = 0; nt < 4; ++nt) {
    const int N = nt * 16 + lm;                       // r3
    for (int mt = 0; mt < 4; ++mt) {
      const int Mrow = (wave * 4 + mt) * 16 + lm;     // i012 (A-frag row)
      v8f acc = vzero;
      #pragma unroll
      for (int kt = 0; kt < 2; ++kt) {
        const int K = kt * 4 + lh * 2;                // i3
        v2f a;  a.x = xrow[Mrow * 8 + K];     a.y = xrow[Mrow * 8 + K + 1];
        v2f bb; bb.x = c3[(N * 8 + K) * 8 + o3];
                bb.y = c3[(N * 8 + K + 1) * 8 + o3];
        acc = wmma4(acc, a, bb);
      }
      const int Mb = (wave * 4 + mt) * 16 + lh * 8;   // C/D row base
      #pragma unroll
      for (int v = 0; v < 8; ++v)
        smem[(Mb + v) * 64 + N] = acc[v];             // T3 flat = i012*64 + r3
    }
  }
  __syncthreads();

  // ---------------- Phase 2: T2[(i0i1)=64][(o2,r2)=512] --------------------
  // A = T3 viewed as (64 x 512): flat = M*512 + K  (K = i2*64 + r3, contiguous)
  // B = W2[K,(o2,r2)] = c2[r2*4096 + i2*512 + o2*64 + r3]
  {
    v8f acc[16];
    #pragma unroll
    for (int i = 0; i < 16; ++i) acc[i] = vzero;

    for (int kt = 0; kt < 128; ++kt) {
      const int Kb = kt * 4 + lh * 2;                 // even
      const int i2 = Kb >> 6, r3 = Kb & 63;           // r3+1 stays in block
      v2f aF[4];
      #pragma unroll
      for (int m = 0; m < 4; ++m) {
        const float* p = &smem[(m * 16 + lm) * 512 + Kb];
        aF[m].x = p[0]; aF[m].y = p[1];
      }
      #pragma unroll
      for (int j = 0; j < 4; ++j) {
        const int N = (wave + 8 * j) * 16 + lm;       // (o2,r2)
        const int o2 = N >> 6, r2 = N & 63;
        const float* q = &c2[r2 * 4096 + i2 * 512 + o2 * 64 + r3];
        v2f bF; bF.x = q[0]; bF.y = q[1];
        #pragma unroll
        for (int m = 0; m < 4; ++m)
          acc[j * 4 + m] = wmma4(acc[j * 4 + m], aF[m], bF);
      }
    }
    __syncthreads();                                  // all T3 reads done
    #pragma unroll
    for (int j = 0; j < 4; ++j) {
      const int N = (wave + 8 * j) * 16 + lm;
      #pragma unroll
      for (int m = 0; m < 4; ++m) {
        const int Mb = m * 16 + lh * 8;
        #pragma unroll
        for (int v = 0; v < 8; ++v)
          smem[(Mb + v) * 512 + N] = acc[j * 4 + m][v]; // T2 flat = (i0i1)*512+(o2,r2)
      }
    }
  }
  __syncthreads();

  // ---------------- Phase 1: T1[(i0,o2)=64][(o1,r1)=512] -------------------
  // A[(i0,o2),(i1,r2)] = T2flat[(i0*8+i1)*512 + o2*64 + r2]
  // B = W1[(i1,r2),(o1,r1)] = c1[r1*4096 + i1*512 + o1*64 + r2]
  {
    v8f acc[16];
    #pragma unroll
    for (int i = 0; i < 16; ++i) acc[i] = vzero;

    for (int kt = 0; kt < 128; ++kt) {
      const int Kb = kt * 4 + lh * 2;
      const int i1 = Kb >> 6, r2 = Kb & 63;
      v2f aF[4];
      #pragma unroll
      for (int m = 0; m < 4; ++m) {
        const int M = m * 16 + lm;                    // (i0,o2)
        const int i0 = M >> 3, o2v = M & 7;
        const float* p = &smem[(i0 * 8 + i1) * 512 + o2v * 64 + r2];
        aF[m].x = p[0]; aF[m].y = p[1];
      }
      #pragma unroll
      for (int j = 0; j < 4; ++j) {
        const int N = (wave + 8 * j) * 16 + lm;       // (o1,r1)
        const int o1 = N >> 6, r1 = N & 63;
        const float* q = &c1[r1 * 4096 + i1 * 512 + o1 * 64 + r2];
        v2f bF; bF.x = q[0]; bF.y = q[1];
        #pragma unroll
        for (int m = 0; m < 4; ++m)
          acc[j * 4 + m] = wmma4(acc[j * 4 + m], aF[m], bF);
      }
    }
    __syncthreads();                                  // all T2 reads done
    #pragma unroll
    for (int j = 0; j < 4; ++j) {
      const int N = (wave + 8 * j) * 16 + lm;
      #pragma unroll
      for (int m = 0; m < 4; ++m) {
        const int Mb = m * 16 + lh * 8;
        #pragma unroll
        for (int v = 0; v < 8; ++v)
          smem[(Mb + v) * 512 + N] = acc[j * 4 + m][v]; // T1 flat = (i0,o2)*512+(o1,r1)
      }
    }
  }
  __syncthreads();

  // ---------------- Phase 0: y[(o1,o2)=64][o0 (pad 16)] --------------------
  // A[(o1,o2),(i0,r1)] = T1flat[(i0*8+o2)*512 + o1*64 + r1]
  // B[(i0,r1),o0] = c0[(i0*8+o0)*64 + r1], zero-padded for o0>=8.
  if (wave < 4) {
    v8f acc = vzero;
    const int M = wave * 16 + lm;                     // (o1,o2)
    const int o1 = M >> 3, o2v = M & 7;
    for (int kt = 0; kt < 128; ++kt) {
      const int Kb = kt * 4 + lh * 2;
      const int i0 = Kb >> 6, r1 = Kb & 63;
      const float* p = &smem[(i0 * 8 + o2v) * 512 + o1 * 64 + r1];
      v2f aF; aF.x = p[0]; aF.y = p[1];
      v2f bF;
      if (lm < 8) {
        const float* q = &c0[(i0 * 8 + lm) * 64 + r1];
        bF.x = q[0]; bF.y = q[1];
      } else {
        bF.x = 0.f;  bF.y = 0.f;
      }
      acc = wmma4(acc, aF, bF);                       // uniform flow, EXEC all-1s
    }
    if (lm < 8) {                                     // valid o0 columns only
      #pragma unroll
      for (int v = 0; v < 8; ++v) {
        const int Mr  = wave * 16 + lh * 8 + v;       // (o1,o2) of D row
        const int po1 = Mr >> 3, po2 = Mr & 7;
        const int out = lm * 512 + po1 * 64 + po2 * 8 + o3;
        y[(size_t)b * 4096 + out] = acc[v] + bias[out];
      }
    }
  }
}

extern "C" void kernel_launch(void* const* d_in, const int* in_sizes, int n_in,
                              void* d_out, int out_size, void* d_ws, size_t ws_size,
                              hipStream_t stream) {
  (void)in_sizes; (void)n_in; (void)d_ws; (void)ws_size; (void)out_size;
  const float* x    = (const float*)d_in[0];
  const float* c0   = (const float*)d_in[1];
  const float* c1   = (const float*)d_in[2];
  const float* c2   = (const float*)d_in[3];
  const float* c3   = (const float*)d_in[4];
  const float* bias = (const float*)d_in[5];
  float* y = (float*)d_out;

  // 1024 batches x 8 o3 values; dynamic LDS = 512*64 f32 = 128 KB (<=320 KB/WGP)
  tt_fused_kernel<<<dim3(8192), dim3(256), 512 * 64 * sizeof(float), stream>>>(
      x, c0, c1, c2, c3, bias, y);
}